// JKNet_9371618640571
// MI455X (gfx1250) — compile-verified
//
#include <hip/hip_runtime.h>

#define N_NODES 100000
#define N_EDGES 640000
#define DIM     128
#define ODIM    40

typedef float v2f __attribute__((ext_vector_type(2)));
typedef float v8f __attribute__((ext_vector_type(8)));

// ---------------- zero fill ----------------
__global__ void zero_kernel(float* __restrict__ p, int n) {
    int i = blockIdx.x * blockDim.x + threadIdx.x;
    int stride = gridDim.x * blockDim.x;
    for (; i < n; i += stride) p[i] = 0.0f;
}

// ---------------- degree count ----------------
__global__ void degree_kernel(const int* __restrict__ dst, float* __restrict__ deg) {
    int e = blockIdx.x * blockDim.x + threadIdx.x;
    if (e < N_EDGES) atomicAdd(&deg[dst[e]], 1.0f);
}

// ---------------- edge scatter-add: agg[dst] += h[src] ----------------
// one wave32 per edge, each lane moves 4 consecutive floats (float4 load)
__global__ void scatter_kernel(const float* __restrict__ h,
                               const int* __restrict__ src,
                               const int* __restrict__ dst,
                               float* __restrict__ agg) {
    int t = blockIdx.x * blockDim.x + threadIdx.x;
    int e = t >> 5;
    int lane = t & 31;
    if (e >= N_EDGES) return;
    int s = src[e];
    int d = dst[e];
    const float4* hv = (const float4*)(h + (size_t)s * DIM);
    float4 v = hv[lane];
    float* a = agg + (size_t)d * DIM + lane * 4;
    atomicAdd(a + 0, v.x);
    atomicAdd(a + 1, v.y);
    atomicAdd(a + 2, v.z);
    atomicAdd(a + 3, v.w);
}

// ---------------- fused SAGE layer: out = act( (agg/deg)@Wl + x@Wr + b ) ----
// 256 threads = 8 waves; block computes 16 rows x 128 cols; wave w -> cols [16w,16w+16)
// A tiles (mean, x) staged in LDS with stride 132 (bank-conflict free for the
// 16-row / K-pair WMMA fragment read pattern).
__launch_bounds__(256)
__global__ void sage_kernel(const float* __restrict__ agg,
                            const float* __restrict__ deg,
                            const float* __restrict__ x,
                            const float* __restrict__ Wl,
                            const float* __restrict__ Wr,
                            const float* __restrict__ bias,   // may be null
                            float* __restrict__ out,
                            int do_relu) {
    __shared__ float sm[16 * 132];
    __shared__ float sx[16 * 132];
    __shared__ float sinv[16];

    const int rowBase = blockIdx.x * 16;
    const int tid = threadIdx.x;

    if (tid < 16) {
        float dg = deg[rowBase + tid];
        sinv[tid] = 1.0f / fmaxf(dg, 1.0f);
    }
    __syncthreads();

    for (int i = tid; i < 16 * 128; i += 256) {
        int r = i >> 7;
        int c = i & 127;
        int g = (rowBase + r) * DIM + c;
        sm[r * 132 + c] = agg[g] * sinv[r];
        sx[r * 132 + c] = x[g];
    }
    __syncthreads();

    const int wave = tid >> 5;
    const int lane = tid & 31;
    const int lrow = lane & 15;   // M index (A row / B col)
    const int kh   = lane >> 4;   // selects K pair {0,1} vs {2,3}
    const int col  = wave * 16 + lrow;

    v8f acc = {};
    for (int k0 = 0; k0 < DIM; k0 += 4) {
        int kk = k0 + 2 * kh;
        v2f am, ax, bl, br;
        am[0] = sm[lrow * 132 + kk];
        am[1] = sm[lrow * 132 + kk + 1];
        ax[0] = sx[lrow * 132 + kk];
        ax[1] = sx[lrow * 132 + kk + 1];
        bl[0] = Wl[kk * DIM + col];
        bl[1] = Wl[(kk + 1) * DIM + col];
        br[0] = Wr[kk * DIM + col];
        br[1] = Wr[(kk + 1) * DIM + col];
        acc = __builtin_amdgcn_wmma_f32_16x16x4_f32(false, am, false, bl,
                                                    (short)0, acc, false, false);
        acc = __builtin_amdgcn_wmma_f32_16x16x4_f32(false, ax, false, br,
                                                    (short)0, acc, false, false);
    }

    float b = bias ? bias[col] : 0.0f;
#pragma unroll
    for (int r = 0; r < 8; ++r) {
        int m = r + 8 * kh;   // C/D layout: VGPR r, lanes 16-31 -> M = r+8
        float v = acc[r] + b;
        if (do_relu) v = fmaxf(v, 0.0f);
        out[(rowBase + m) * DIM + col] = v;
    }
}

// ---------------- classifier: out = concat(h0,h1,h2) @ Wc + bc --------------
// 8 waves per block, one 16-row tile per wave; 3 column tiles (48 cols, 40 valid)
__launch_bounds__(256)
__global__ void cls_kernel(const float* __restrict__ h0,
                           const float* __restrict__ h1,
                           const float* __restrict__ h2,
                           const float* __restrict__ Wc,
                           const float* __restrict__ bc,
                           float* __restrict__ out) {
    const int wave = threadIdx.x >> 5;
    const int lane = threadIdx.x & 31;
    const int rowBase = (blockIdx.x * 8 + wave) * 16;
    if (rowBase >= N_NODES) return;        // wave-uniform: EXEC stays all-1s
    const int lrow = lane & 15;
    const int kh   = lane >> 4;

    const float* hs[3] = { h0, h1, h2 };

    for (int ct = 0; ct < 3; ++ct) {
        const int col = ct * 16 + lrow;
        const bool cok = (col < ODIM);
        v8f acc = {};
        for (int k0 = 0; k0 < 3 * DIM; k0 += 4) {
            int kk = k0 + 2 * kh;
            const float* hsrc = hs[kk >> 7];   // kk, kk+1 in same 128-chunk
            int kl = kk & 127;
            v2f a, b;
            a[0] = hsrc[(rowBase + lrow) * DIM + kl];
            a[1] = hsrc[(rowBase + lrow) * DIM + kl + 1];
            b[0] = cok ? Wc[kk * ODIM + col] : 0.0f;         // predicated load
            b[1] = cok ? Wc[(kk + 1) * ODIM + col] : 0.0f;
            acc = __builtin_amdgcn_wmma_f32_16x16x4_f32(false, a, false, b,
                                                        (short)0, acc, false, false);
        }
        if (cok) {
            float bb = bc[col];
#pragma unroll
            for (int r = 0; r < 8; ++r) {
                int m = r + 8 * kh;
                out[(rowBase + m) * ODIM + col] = acc[r] + bb;
            }
        }
    }
}

// ---------------- launcher ----------------
extern "C" void kernel_launch(void* const* d_in, const int* in_sizes, int n_in,
                              void* d_out, int out_size, void* d_ws, size_t ws_size,
                              hipStream_t stream) {
    const float* x   = (const float*)d_in[0];
    const int*   ei  = (const int*)  d_in[1];
    const float* Wl0 = (const float*)d_in[2];
    const float* Wr0 = (const float*)d_in[3];
    const float* Wl1 = (const float*)d_in[4];
    const float* Wr1 = (const float*)d_in[5];
    const float* Wl2 = (const float*)d_in[6];
    const float* Wr2 = (const float*)d_in[7];
    const float* b2  = (const float*)d_in[8];
    const float* Wc  = (const float*)d_in[9];
    const float* bc  = (const float*)d_in[10];
    float* out = (float*)d_out;

    const int* src = ei;
    const int* dst = ei + N_EDGES;

    float* ws  = (float*)d_ws;
    float* deg = ws;                              // 100000 floats (pad to 102400)
    float* agg = ws + 102400;                     // N*DIM, reused every layer
    float* h0  = agg + (size_t)N_NODES * DIM;
    float* h1  = h0  + (size_t)N_NODES * DIM;
    float* h2  = h1  + (size_t)N_NODES * DIM;

    const int ZB = 4096, ZT = 256;                // grid-stride zero fill
    const int featN = N_NODES * DIM;

    // degree (shared by all layers)
    zero_kernel<<<ZB, ZT, 0, stream>>>(deg, N_NODES);
    degree_kernel<<<(N_EDGES + 255) / 256, 256, 0, stream>>>(dst, deg);

    const int scatterBlocks = (N_EDGES * 32) / 256;   // one wave per edge
    const int sageBlocks    = N_NODES / 16;           // 6250, exact
    const int clsBlocks     = (N_NODES / 16 + 7) / 8; // 782

    // layer 0: h0 = relu(mean(x)@Wl0 + x@Wr0)
    zero_kernel<<<ZB, ZT, 0, stream>>>(agg, featN);
    scatter_kernel<<<scatterBlocks, 256, 0, stream>>>(x, src, dst, agg);
    sage_kernel<<<sageBlocks, 256, 0, stream>>>(agg, deg, x, Wl0, Wr0, nullptr, h0, 1);

    // layer 1: h1 = relu(mean(h0)@Wl1 + h0@Wr1)
    zero_kernel<<<ZB, ZT, 0, stream>>>(agg, featN);
    scatter_kernel<<<scatterBlocks, 256, 0, stream>>>(h0, src, dst, agg);
    sage_kernel<<<sageBlocks, 256, 0, stream>>>(agg, deg, h0, Wl1, Wr1, nullptr, h1, 1);

    // layer 2: h2 = mean(h1)@Wl2 + h1@Wr2 + b2   (no relu)
    zero_kernel<<<ZB, ZT, 0, stream>>>(agg, featN);
    scatter_kernel<<<scatterBlocks, 256, 0, stream>>>(h1, src, dst, agg);
    sage_kernel<<<sageBlocks, 256, 0, stream>>>(agg, deg, h1, Wl2, Wr2, b2, h2, 0);

    // classifier over JK-concat
    cls_kernel<<<clsBlocks, 256, 0, stream>>>(h0, h1, h2, Wc, bc, out);
}